// LearnableGaborBlur_76029511074373
// MI455X (gfx1250) — compile-verified
//
#include <hip/hip_runtime.h>
#include <math.h>

typedef float v2f __attribute__((ext_vector_type(2)));
typedef float v8f __attribute__((ext_vector_type(8)));

#define KS      7
#define HALF    3
#define TILE_Y  16
#define TILE_X  128
#define SROWS   22      // TILE_Y + 6 halo rows
#define SCOLS   136     // TILE_X + 8 (halo 3 left, 5 right incl. dead band cols)
#define SSTR    136
#define SOSTR   132     // output transpose stride (pad vs 64 banks)
#define NCH     6       // K-chunks of 4 covering k = 0..23 (band needs k <= 21)

// ---------------- per-channel Gabor kernel synthesis ----------------
__global__ __launch_bounds__(256) void gabor_weights_kernel(
    const float* __restrict__ log_sigma, const float* __restrict__ log_freq,
    const float* __restrict__ theta, float* __restrict__ w, int C)
{
    int c = blockIdx.x * blockDim.x + threadIdx.x;
    if (c >= C) return;
    float sigma = expf(log_sigma[c]);
    float freq  = expf(log_freq[c]);
    float ct = cosf(theta[c]);
    float st = sinf(theta[c]);
    float k[KS * KS];
    float sum = 0.f;
    for (int i = 0; i < KS; ++i) {          // i = dy axis (meshgrid 'ij' first axis)
        float gx = (float)(i - HALF);
        for (int j = 0; j < KS; ++j) {      // j = dx axis
            float gy = (float)(j - HALF);
            float u = ( gx * ct + gy * st) / sigma;
            float v = (-gx * st + gy * ct) / sigma;
            float g = expf(-0.5f * (u * u + v * v));
            float val = g * cosf(2.0f * (float)M_PI * freq * u);
            k[i * KS + j] = val;
            sum += val;
        }
    }
    float inv = 1.0f / sum;
    for (int j = 0; j < KS * KS; ++j) w[c * KS * KS + j] = k[j] * inv;
}

// ---------------- depthwise conv via V_WMMA_F32_16X16X4_F32 ----------------
__global__ __launch_bounds__(256) void gabor_conv_kernel(
    const float* __restrict__ x, const float* __restrict__ w,
    float* __restrict__ out, int H, int W, int C)
{
    __shared__ __align__(16) float sInp[SROWS * SSTR];         // input tile + halo
    __shared__ __align__(16) float sOut[TILE_Y * SOSTR];       // output transpose
    __shared__ __align__(16) float sA[KS * NCH * 64];          // pre-laid-out A operands
    __shared__ float wds[KS * KS];

    const int tid = threadIdx.x;
    const int bc  = blockIdx.z;
    const int c   = bc % C;
    const size_t base = (size_t)bc * (size_t)H * (size_t)W;
    const int y0 = blockIdx.y * TILE_Y;
    const int xb = blockIdx.x * TILE_X;

    // ---- phase 1: weights -> LDS, zero-padded input tile -> LDS ----
    if (tid < KS * KS) wds[tid] = w[c * KS * KS + tid];

    for (int i = tid; i < SROWS * SCOLS; i += 256) {
        int r   = i / SCOLS;
        int col = i - r * SCOLS;
        int gy = y0 - HALF + r;
        int gx = xb - HALF + col;
        float v = 0.f;
        if (gy >= 0 && gy < H && gx >= 0 && gx < W)
            v = x[base + (size_t)gy * W + gx];
        sInp[r * SSTR + col] = v;
    }
    __syncthreads();

    // ---- phase 2: build all A operands in WMMA lane layout (branch-free hot loop) ----
    // A_dy[m, kk] = w[dy, kk - m] band; lane l<16 holds K=kk0,kk0+1 ; l>=16 holds K=kk0+2,kk0+3
    for (int idx = tid; idx < KS * NCH * 64; idx += 256) {
        int pair = idx >> 6;              // (dy*NCH + ch)
        int e    = idx & 63;              // lane*2 + p
        int dy = pair / NCH;
        int ch = pair - dy * NCH;
        int l  = e >> 1;
        int p  = e & 1;
        int m  = l & 15;
        int hh = l >> 4;
        int kk = ch * 4 + 2 * hh + p;
        int i  = kk - m;                  // tap index dx
        sA[idx] = (i >= 0 && i < KS) ? wds[dy * KS + i] : 0.f;
    }
    __syncthreads();

    // ---- phase 3: 42 chained WMMAs, 2 ds_load_b64 per WMMA, double-buffered ----
    const int lane = tid & 31;
    const int wid  = tid >> 5;            // wave id 0..7 (wave32)
    const int n    = lane & 15;           // M index for A, N index for B/C
    const int h    = lane >> 4;           // K-half selector
    const int xloc = wid * 16;

    const float* aBase = &sA[lane * 2];
    const float* bBase = &sInp[n * SSTR + xloc + 2 * h];

    v8f acc = {0.f, 0.f, 0.f, 0.f, 0.f, 0.f, 0.f, 0.f};

    v2f a[2][NCH], b[2][NCH];
    #pragma unroll
    for (int ch = 0; ch < NCH; ++ch) {
        a[0][ch] = *(const v2f*)(aBase + ch * 64);
        b[0][ch] = *(const v2f*)(bBase + ch * 4);
    }
    #pragma unroll
    for (int dy = 0; dy < KS; ++dy) {
        const int cur = dy & 1;
        const int nxt = cur ^ 1;
        if (dy < KS - 1) {
            #pragma unroll
            for (int ch = 0; ch < NCH; ++ch) {
                a[nxt][ch] = *(const v2f*)(aBase + ((dy + 1) * NCH + ch) * 64);
                b[nxt][ch] = *(const v2f*)(bBase + (dy + 1) * SSTR + ch * 4);
            }
        }
        #pragma unroll
        for (int ch = 0; ch < NCH; ++ch)
            acc = __builtin_amdgcn_wmma_f32_16x16x4_f32(
                false, a[cur][ch], false, b[cur][ch], (short)0, acc, false, false);
    }

    // ---- phase 4: C/D (VGPR i -> M=i+8h, N=lane&15) -> LDS transpose -> b128 stores ----
    #pragma unroll
    for (int i = 0; i < 8; ++i)
        sOut[n * SOSTR + xloc + h * 8 + i] = acc[i];
    __syncthreads();

    #pragma unroll
    for (int p = 0; p < 2; ++p) {
        int idx = tid + p * 256;
        int row = idx >> 5;
        int q   = idx & 31;
        float4 v = *(const float4*)(&sOut[row * SOSTR + q * 4]);
        *(float4*)(&out[base + (size_t)(y0 + row) * W + xb + q * 4]) = v;
    }
}

extern "C" void kernel_launch(void* const* d_in, const int* in_sizes, int n_in,
                              void* d_out, int out_size, void* d_ws, size_t ws_size,
                              hipStream_t stream) {
    const float* x  = (const float*)d_in[0];
    const float* ls = (const float*)d_in[1];
    const float* lf = (const float*)d_in[2];
    const float* th = (const float*)d_in[3];
    float* out = (float*)d_out;
    float* wbuf = (float*)d_ws;           // C*49 floats

    const int C = in_sizes[1];
    const int H = 256, W = 256;
    const int BC = (int)((long long)in_sizes[0] / ((long long)H * W));

    gabor_weights_kernel<<<(C + 255) / 256, 256, 0, stream>>>(ls, lf, th, wbuf, C);

    dim3 grid(W / TILE_X, H / TILE_Y, BC);
    gabor_conv_kernel<<<grid, 256, 0, stream>>>(x, wbuf, out, H, W, C);
}